// RelationGAT_1906965479723
// MI455X (gfx1250) — compile-verified
//
#include <hip/hip_runtime.h>
#include <math.h>

typedef float v2f __attribute__((ext_vector_type(2)));
typedef float v8f __attribute__((ext_vector_type(8)));

#define D 128  // D_IN == D_OUT == 128

// ---------------------------------------------------------------------------
// Kernel A: fuse weights.
//   M1[d',d] = sum_o Wq[o,d'] * Wk[o,d]   (= Wq^T @ Wk)   -> used for scores
//   M2[j ,d] = sum_o Wv[j,o ] * Wk[o,d]   (= Wv  @ Wk)    -> used for output
// 2 * 128*128 elements, 128-long dots: negligible cost.
// ---------------------------------------------------------------------------
__global__ void fuse_weights_kernel(const float* __restrict__ Wq,
                                    const float* __restrict__ Wk,
                                    const float* __restrict__ Wv,
                                    float* __restrict__ M1,
                                    float* __restrict__ M2) {
  int t = blockIdx.x * blockDim.x + threadIdx.x;
  if (t < D * D) {
    int dp = t >> 7, e = t & (D - 1);
    float s = 0.f;
    #pragma unroll 8
    for (int o = 0; o < D; ++o) s = fmaf(Wq[o * D + dp], Wk[o * D + e], s);
    M1[t] = s;
  } else {
    int t2 = t - D * D;
    int j = t2 >> 7, e = t2 & (D - 1);
    float s = 0.f;
    #pragma unroll 8
    for (int o = 0; o < D; ++o) s = fmaf(Wv[j * D + o], Wk[o * D + e], s);
    M2[t2] = s;
  }
}

// ---------------------------------------------------------------------------
// Kernel B: C[m,n] = sum_k A[m,k] * Bm[k,n]   (Bm row-major [K x N], K=N=128)
// Full-precision f32 WMMA 16x16x4. Block = 256 threads = 8 waves; each wave
// owns one 16x16 tile of C (wave index = column tile), block row = 16 rows.
//
// A 16x4 f32 fragment (ISA 7.12.2): lanes 0-15 -> M=lane, VGPR0=K0,VGPR1=K1;
// lanes 16-31 -> K2,K3.  => per-lane float2 at A[row, 4*kb + 2*half].
// B 4x16 fragment mirrors it: per-lane rows K = 4*kb + 2*half + {0,1}, col n.
// C/D: VGPR i -> row (i + 8*half), col (lane & 15).
// ---------------------------------------------------------------------------
__global__ void gemm_xB_kernel(const float* __restrict__ A,
                               const float* __restrict__ Bm,
                               float* __restrict__ C) {
  const int lane  = threadIdx.x & 31;
  const int wave  = threadIdx.x >> 5;       // 0..7 -> N tile
  const int half  = lane >> 4;
  const int lm    = lane & 15;
  const int mbase = blockIdx.x * 16;
  const int nbase = wave * 16;

  v8f c = {};
  const float* arow = A + (size_t)(mbase + lm) * D + 2 * half;
  const float* bcol = Bm + (size_t)(2 * half) * D + (nbase + lm);

  #pragma unroll 4
  for (int kb = 0; kb < D / 4; ++kb) {
    v2f a = *(const v2f*)(arow + 4 * kb);                 // global_load_b64
    v2f b;
    b.x = bcol[(size_t)(4 * kb) * D];                     // strided b32
    b.y = bcol[(size_t)(4 * kb) * D + D];
    c = __builtin_amdgcn_wmma_f32_16x16x4_f32(false, a, false, b,
                                              (short)0, c, false, false);
  }

  float* crow = C + (size_t)(mbase + 8 * half) * D + nbase + lm;
  #pragma unroll
  for (int i = 0; i < 8; ++i) crow[(size_t)i * D] = c[i];
}

// ---------------------------------------------------------------------------
// Kernel D: C[m,n] = sum_k A[m,k] * Bt[n,k]  (Bt row-major [N x K]; i.e. B^T)
// Same structure; B fragment is a contiguous float2 from Bt's row n.
// ---------------------------------------------------------------------------
__global__ void gemm_xBt_kernel(const float* __restrict__ A,
                                const float* __restrict__ Bt,
                                float* __restrict__ C) {
  const int lane  = threadIdx.x & 31;
  const int wave  = threadIdx.x >> 5;
  const int half  = lane >> 4;
  const int lm    = lane & 15;
  const int mbase = blockIdx.x * 16;
  const int nbase = wave * 16;

  v8f c = {};
  const float* arow = A + (size_t)(mbase + lm) * D + 2 * half;
  const float* brow = Bt + (size_t)(nbase + lm) * D + 2 * half;

  #pragma unroll 4
  for (int kb = 0; kb < D / 4; ++kb) {
    v2f a = *(const v2f*)(arow + 4 * kb);                 // global_load_b64
    v2f b = *(const v2f*)(brow + 4 * kb);                 // global_load_b64
    c = __builtin_amdgcn_wmma_f32_16x16x4_f32(false, a, false, b,
                                              (short)0, c, false, false);
  }

  float* crow = C + (size_t)(mbase + 8 * half) * D + nbase + lm;
  #pragma unroll
  for (int i = 0; i < 8; ++i) crow[(size_t)i * D] = c[i];
}

// ---------------------------------------------------------------------------
// Kernel C: streaming attention with online softmax.
// One wave per batch row b; lane owns dims [4*lane .. 4*lane+3].
// neighbor[b,n,:] is loaded ONCE (b128 per lane), used for both the score
// dot-product (cross-lane shfl_xor reduction) and the attn-weighted sum.
//   scores[b,n] = neighbor[b,n,:] . qk[b,:]
//   nbar[b,:]   = softmax_n(scores) @ neighbor[b]
// ---------------------------------------------------------------------------
__global__ void attn_stream_kernel(const float* __restrict__ neighbor,
                                   const float* __restrict__ qk,
                                   float* __restrict__ nbar,
                                   int Bn, int N) {
  const int lane = threadIdx.x & 31;
  const int wave = threadIdx.x >> 5;
  const int b = blockIdx.x * (blockDim.x >> 5) + wave;
  if (b >= Bn) return;

  const float4 qv = *(const float4*)(qk + (size_t)b * D + 4 * lane);
  const float* nb_base = neighbor + (size_t)b * N * D + 4 * lane;

  float m = -INFINITY;
  float l = 0.f;
  float4 acc = {0.f, 0.f, 0.f, 0.f};

  for (int n = 0; n < N; ++n) {
    if (n + 8 < N)  // speculative stream prefetch (global_prefetch_b8)
      __builtin_prefetch(nb_base + (size_t)(n + 8) * D, 0, 1);

    float4 v = *(const float4*)(nb_base + (size_t)n * D);  // global_load_b128

    // per-lane partial dot over 4 dims, then wave32 butterfly reduce
    float s = qv.x * v.x;
    s = fmaf(qv.y, v.y, s);
    s = fmaf(qv.z, v.z, s);
    s = fmaf(qv.w, v.w, s);
    s += __shfl_xor(s, 16);
    s += __shfl_xor(s, 8);
    s += __shfl_xor(s, 4);
    s += __shfl_xor(s, 2);
    s += __shfl_xor(s, 1);

    // online softmax update
    float mn   = fmaxf(m, s);
    float csc  = __expf(m - mn);   // first iter: exp(-inf) = 0
    float p    = __expf(s - mn);
    l = fmaf(l, csc, p);
    acc.x = fmaf(acc.x, csc, p * v.x);
    acc.y = fmaf(acc.y, csc, p * v.y);
    acc.z = fmaf(acc.z, csc, p * v.z);
    acc.w = fmaf(acc.w, csc, p * v.w);
    m = mn;
  }

  const float inv = 1.0f / l;
  float4 o = {acc.x * inv, acc.y * inv, acc.z * inv, acc.w * inv};
  *(float4*)(nbar + (size_t)b * D + 4 * lane) = o;
}

// ---------------------------------------------------------------------------
// Launch: M1/M2 -> qk = x@M1 (WMMA) -> nbar (streaming) -> out = nbar@M2^T (WMMA)
// Workspace layout (f32): M1[128*128] | M2[128*128] | qk[B*128] | nbar[B*128]
// ---------------------------------------------------------------------------
extern "C" void kernel_launch(void* const* d_in, const int* in_sizes, int n_in,
                              void* d_out, int out_size, void* d_ws, size_t ws_size,
                              hipStream_t stream) {
  (void)n_in; (void)out_size; (void)ws_size;

  const float* x        = (const float*)d_in[0];
  const float* neighbor = (const float*)d_in[1];
  const float* Wq       = (const float*)d_in[2];
  const float* Wk       = (const float*)d_in[3];
  const float* Wv       = (const float*)d_in[4];
  float*       out      = (float*)d_out;

  const int Bn = in_sizes[0] / D;             // 16384
  const int N  = in_sizes[1] / (Bn * D);      // 50

  float* M1   = (float*)d_ws;
  float* M2   = M1 + D * D;
  float* qk   = M2 + D * D;
  float* nbar = qk + (size_t)Bn * D;

  fuse_weights_kernel<<<(2 * D * D) / 256, 256, 0, stream>>>(Wq, Wk, Wv, M1, M2);
  gemm_xB_kernel<<<Bn / 16, 256, 0, stream>>>(x, M1, qk);
  attn_stream_kernel<<<(Bn + 7) / 8, 256, 0, stream>>>(neighbor, qk, nbar, Bn, N);
  gemm_xBt_kernel<<<Bn / 16, 256, 0, stream>>>(nbar, M2, out);
}